// My_GATLayer_10015863734677
// MI455X (gfx1250) — compile-verified
//
#include <hip/hip_runtime.h>
#include <hip/hip_bf16.h>

typedef float v2f __attribute__((ext_vector_type(2)));
typedef float v8f __attribute__((ext_vector_type(8)));

// ---------------------------------------------------------------------------
// Kernel 1: fp32 WMMA projections.  h_s = h @ W_self^T, z = h @ W_func^T.
// Block = 256 threads = 8 waves; one block handles a 16-node tile.
//   wave 0..3 -> h_s col-tiles 0..3, wave 4..7 -> z col-tiles 0..3.
// Each wave: 16x16 f32 accumulator, 16 x V_WMMA_F32_16X16X4_F32 over K=64.
// ---------------------------------------------------------------------------
__global__ __launch_bounds__(256) void gat_proj_wmma(
    const float* __restrict__ h,      // [N,64]
    const float* __restrict__ Wself,  // [64,64] row-major
    const float* __restrict__ Wfunc,  // [64,64] row-major
    float* __restrict__ h_s,          // [N,64]
    float* __restrict__ z,            // [N,64]
    int nNodes)
{
    const int wave = threadIdx.x >> 5;
    const int lane = threadIdx.x & 31;
    const int nodeBase = blockIdx.x * 16;

    const float* __restrict__ W = (wave < 4) ? Wself : Wfunc;
    float* __restrict__ out     = (wave < 4) ? h_s : z;
    const int nt = (wave & 3) * 16;      // output column tile base

    const int m     = lane & 15;         // A row / B col / D col within tile
    const int khalf = lane >> 4;         // 0 or 1: which K pair this lane holds

    int arow_i = nodeBase + m;
    if (arow_i >= nNodes) arow_i = nNodes - 1;         // clamp (reads only)
    const float* __restrict__ arow = h + (size_t)arow_i * 64 + 2 * khalf;
    const float* __restrict__ brow = W + (size_t)(nt + m) * 64 + 2 * khalf;

    v8f acc = {};
#pragma unroll
    for (int ko = 0; ko < 64; ko += 4) {
        v2f a = *(const v2f*)(arow + ko);   // A[m][ko+2*khalf .. +1]
        v2f b = *(const v2f*)(brow + ko);   // B[k][n] = W[nt+n][k]
        acc = __builtin_amdgcn_wmma_f32_16x16x4_f32(
            /*neg_a=*/false, a, /*neg_b=*/false, b,
            /*c_mod=*/(short)0, acc, /*reuse_a=*/false, /*reuse_b=*/false);
    }

    // D layout: lane holds col n = lane&15; VGPR r holds row r + 8*khalf.
    const int n = lane & 15;
#pragma unroll
    for (int r = 0; r < 8; ++r) {
        const int mm = r + 8 * khalf;
        const int row = nodeBase + mm;
        if (row < nNodes)
            out[(size_t)row * 64 + nt + n] = acc[r];
    }
}

// ---------------------------------------------------------------------------
// Kernel 2: attention logits per node: a_l = z . Wa[:64], a_r = z . Wa[64:].
// ---------------------------------------------------------------------------
__global__ __launch_bounds__(256) void gat_attn_logits(
    const float* __restrict__ z, const float* __restrict__ Wattn,
    float* __restrict__ a_l, float* __restrict__ a_r, int nNodes)
{
    const int n = blockIdx.x * blockDim.x + threadIdx.x;
    if (n >= nNodes) return;
    const float* __restrict__ zr = z + (size_t)n * 64;
    float sl = 0.f, sr = 0.f;
#pragma unroll
    for (int j = 0; j < 64; ++j) {
        const float v = zr[j];
        sl += v * Wattn[j];
        sr += v * Wattn[64 + j];
    }
    a_l[n] = sl;
    a_r[n] = sr;
}

// Monotonic float <-> uint key for atomic max over signed floats.
__device__ __forceinline__ unsigned f2key(float f) {
    unsigned b = __float_as_uint(f);
    return (b & 0x80000000u) ? ~b : (b | 0x80000000u);
}
__device__ __forceinline__ float key2f(unsigned k) {
    return __uint_as_float((k & 0x80000000u) ? (k & 0x7FFFFFFFu) : ~k);
}

// ---------------------------------------------------------------------------
// Kernel 3: edge logits + segment max (atomicMax on monotone uint key).
// ---------------------------------------------------------------------------
__global__ __launch_bounds__(256) void gat_edge_max(
    const float* __restrict__ a_l, const float* __restrict__ a_r,
    const int* __restrict__ src, const int* __restrict__ dst,
    float* __restrict__ e_buf, unsigned* __restrict__ emax_key, int nEdges)
{
    const int e = blockIdx.x * blockDim.x + threadIdx.x;
    if (e >= nEdges) return;
    const int d = dst[e];
    float v = a_l[src[e]] + a_r[d];
    v = (v > 0.f) ? v : 0.01f * v;        // leaky_relu(0.01)
    e_buf[e] = v;
    atomicMax(emax_key + d, f2key(v));
}

// ---------------------------------------------------------------------------
// Kernel 4: e_exp = exp(e - max[dst]); denom[dst] += e_exp (fp32 atomics).
// ---------------------------------------------------------------------------
__global__ __launch_bounds__(256) void gat_edge_exp(
    const int* __restrict__ dst, const unsigned* __restrict__ emax_key,
    float* __restrict__ e_buf, float* __restrict__ denom, int nEdges)
{
    const int e = blockIdx.x * blockDim.x + threadIdx.x;
    if (e >= nEdges) return;
    const int d = dst[e];
    const float w = __expf(e_buf[e] - key2f(emax_key[d]));
    e_buf[e] = w;
    atomicAdd(denom + d, w);
}

// ---------------------------------------------------------------------------
// Kernel 5: agg[dst] += (e_exp/denom[dst]) * z[src].  64 threads per edge:
// contiguous 256B gather of z[src] and contiguous fp32 atomic adds.
// ---------------------------------------------------------------------------
__global__ __launch_bounds__(256) void gat_aggregate(
    const int* __restrict__ src, const int* __restrict__ dst,
    const float* __restrict__ e_buf, const float* __restrict__ denom,
    const float* __restrict__ z, float* __restrict__ agg, int nEdges)
{
    const long long t = (long long)blockIdx.x * 256 + threadIdx.x;
    const int e = (int)(t >> 6);
    const int j = (int)(t & 63);
    if (e >= nEdges) return;
    const int s = src[e], d = dst[e];
    const float a = e_buf[e] / denom[d];
    atomicAdd(agg + (size_t)d * 64 + j, a * z[(size_t)s * 64 + j]);
}

// ---------------------------------------------------------------------------
// Kernel 6: out = h + relu(h_s + agg)
// ---------------------------------------------------------------------------
__global__ __launch_bounds__(256) void gat_finalize(
    const float* __restrict__ h, const float* __restrict__ h_s,
    const float* __restrict__ agg, float* __restrict__ out, int total)
{
    const int i = blockIdx.x * blockDim.x + threadIdx.x;
    if (i >= total) return;
    const float hn = h_s[i] + agg[i];
    out[i] = h[i] + (hn > 0.f ? hn : 0.f);
}

extern "C" void kernel_launch(void* const* d_in, const int* in_sizes, int n_in,
                              void* d_out, int out_size, void* d_ws, size_t ws_size,
                              hipStream_t stream)
{
    const float* h      = (const float*)d_in[0];
    const int*   src    = (const int*)  d_in[1];
    const int*   dst    = (const int*)  d_in[2];
    const float* W_self = (const float*)d_in[3];
    const float* W_func = (const float*)d_in[4];
    const float* W_attn = (const float*)d_in[5];
    float* out = (float*)d_out;

    const int nNodes = in_sizes[0] / 64;
    const int nEdges = in_sizes[1];
    const size_t fN = (size_t)nNodes * 64;

    // Workspace carve-up (floats unless noted).
    float*    hs    = (float*)d_ws;            // [N,64]
    float*    zbuf  = hs + fN;                 // [N,64]
    float*    al    = zbuf + fN;               // [N]
    float*    ar    = al + nNodes;             // [N]
    unsigned* emax  = (unsigned*)(ar + nNodes);// [N]  (uint keys)
    float*    denom = (float*)(emax + nNodes); // [N]
    float*    ebuf  = denom + nNodes;          // [E]
    float*    agg   = ebuf + nEdges;           // [N,64]

    // Zero the accumulation buffers (emax & denom are contiguous).
    hipMemsetAsync(emax, 0, (size_t)2 * nNodes * sizeof(float), stream);
    hipMemsetAsync(agg, 0, fN * sizeof(float), stream);

    const int nodeTiles = (nNodes + 15) / 16;
    gat_proj_wmma<<<nodeTiles, 256, 0, stream>>>(h, W_self, W_func, hs, zbuf, nNodes);

    gat_attn_logits<<<(nNodes + 255) / 256, 256, 0, stream>>>(zbuf, W_attn, al, ar, nNodes);

    gat_edge_max<<<(nEdges + 255) / 256, 256, 0, stream>>>(al, ar, src, dst, ebuf, emax, nEdges);

    gat_edge_exp<<<(nEdges + 255) / 256, 256, 0, stream>>>(dst, emax, ebuf, denom, nEdges);

    const long long aggThreads = (long long)nEdges * 64;
    gat_aggregate<<<(unsigned)((aggThreads + 255) / 256), 256, 0, stream>>>(
        src, dst, ebuf, denom, zbuf, agg, nEdges);

    const int total = nNodes * 64;
    gat_finalize<<<(total + 255) / 256, 256, 0, stream>>>(h, hs, agg, out, total);
}